// LB_CNN_34840774705398
// MI455X (gfx1250) — compile-verified
//
#include <hip/hip_runtime.h>
#include <hip/hip_bf16.h>

// ---------------------------------------------------------------------------
// Problem constants (from reference): B=4096, F=2048, M=14, C=100, K=2^14.
// Factorization: k = hi*128 + lo (7 bits each).
//   exp(cb[k]·le_b - S_b) = EA[b,hi]*EB[b,lo]
//   denom_b   = EA_b^T V_c EB_b            V_c  = pi[c] as 128x128 (hi major)
//   EIIterm_b = EA_b^T (V_c .* LT) EB_b    LT[k]= logsumexp_c linear_pi
//   N_j (hi bits) from u = V_c EB_b ; N_j (lo bits) from v = V_c^T EA_b
// Class-grouped -> fp32 WMMA GEMMs (V_WMMA_F32_16X16X4_F32).
// Roofline: dominated by streaming x_in (32 MB @ 23.3 TB/s ~ 1.4 us); all
// other arrays (<= 6.4 MB) are L2-resident on the 192 MB L2.
// ---------------------------------------------------------------------------

#define BB 4096
#define FF 2048
#define MM 14
#define CC 100
#define KK 16384
#define LEP 16          // le padded columns
#define KSPLIT 16       // split-K factor for the le GEMM

typedef __attribute__((ext_vector_type(2))) float v2f;
typedef __attribute__((ext_vector_type(8))) float v8f;

static __device__ __forceinline__ v8f wmma_f32_16x16x4(v2f a, v2f b, v8f c) {
  // 8-arg form: (neg_a, A, neg_b, B, c_mod, C, reuse_a, reuse_b)
  return __builtin_amdgcn_wmma_f32_16x16x4_f32(false, a, false, b, (short)0, c,
                                               false, false);
}

static __device__ __forceinline__ float softplusf(float x) {
  return (x > 0.f) ? (x + log1pf(expf(-x))) : log1pf(expf(x));
}

// --------------------------- stage 0: init --------------------------------
__global__ void k_init(float* __restrict__ le_pad, const float* __restrict__ alpha_0,
                       int* __restrict__ counts, float* __restrict__ out) {
  int t = blockIdx.x * blockDim.x + threadIdx.x;
  if (t < BB * LEP) {
    int j = t & (LEP - 1);
    le_pad[t] = (j < MM) ? alpha_0[j] : 0.f;
  }
  if (t < CC) counts[t] = 0;
  if (t == 0) out[0] = 0.f;
}

// ------------------- stage 1: le = alpha_0 + x @ alpha^T -------------------
// WMMA f32 16x16x4, one wave per (16-row tile, 128-wide K chunk), split-K via
// atomic f32 adds. Both operands staged in LDS: inner loop is ds_load + wmma
// only (no predication, no global loads) so the kernel runs at the x_in
// streaming roofline.
__global__ void __launch_bounds__(32)
k_le_gemm(const float* __restrict__ x, const float* __restrict__ alpha,
          float* __restrict__ le_pad) {
  __shared__ __align__(16) float xs[16 * 132];  // x tile, padded rows (16B aligned)
  __shared__ __align__(16) float at[128 * 16];  // alpha^T chunk [k][n], n>=14 -> 0
  const int m0 = blockIdx.x * 16;
  const int k0 = blockIdx.y * (FF / KSPLIT);    // 128-wide K chunk
  const int lane = threadIdx.x;                 // wave32

  // x tile: 16 rows x 128 cols as float4 (global_load_b128, coalesced)
  for (int i = 0; i < 16; ++i) {
    int e = lane + i * 32;                      // float4 index, 512 total
    int r = e >> 5, c4 = e & 31;
    float4 v = *(const float4*)(x + (size_t)(m0 + r) * FF + k0 + c4 * 4);
    *(float4*)(xs + r * 132 + c4 * 4) = v;
  }
  // alpha^T chunk: at[k][n] = alpha[n*F + k0 + k], zero-padded n >= 14
  for (int i = 0; i < 64; ++i) {
    int e = lane + i * 32;                      // e = n*128 + k (coalesced in k)
    int n = e >> 7, k = e & 127;
    at[k * 16 + n] = (n < MM) ? alpha[(size_t)n * FF + k0 + k] : 0.f;
  }
  __syncthreads();

  const int mrow = lane & 15;
  const int off0 = (lane < 16) ? 0 : 2;
  const int ncol = lane & 15;                   // output column (padded M dim)
  v8f acc = {0.f, 0.f, 0.f, 0.f, 0.f, 0.f, 0.f, 0.f};
#pragma unroll 4
  for (int kk = 0; kk < 128; kk += 4) {
    v2f a, b;
    a.x = xs[mrow * 132 + kk + off0];
    a.y = xs[mrow * 132 + kk + off0 + 1];
    b.x = at[(kk + off0) * 16 + ncol];
    b.y = at[(kk + off0 + 1) * 16 + ncol];
    acc = wmma_f32_16x16x4(a, b, acc);
  }
#pragma unroll
  for (int r = 0; r < 8; ++r) {
    int mr = (lane < 16) ? r : (r + 8);
    atomicAdd(&le_pad[(size_t)(m0 + mr) * LEP + ncol], acc[r]);
  }
}

// --------------- stage 2: EA/EB tables + EIV accumulation ------------------
__global__ void __launch_bounds__(128)
k_eab(const float* __restrict__ le_pad, float* __restrict__ EA,
      float* __restrict__ EB, float* __restrict__ out) {
  const int b = blockIdx.x;
  const int t = threadIdx.x;   // 0..127 == hi (and lo) index
  float le[MM];
#pragma unroll
  for (int j = 0; j < MM; ++j) le[j] = le_pad[(size_t)b * LEP + j];
  float shi = 0.f, slo = 0.f;
#pragma unroll
  for (int j = 0; j < 7; ++j) shi += softplusf(le[j]);
#pragma unroll
  for (int j = 7; j < MM; ++j) slo += softplusf(le[j]);

  float da = 0.f, db = 0.f;
#pragma unroll
  for (int j = 0; j < 7; ++j) {
    if ((t >> (6 - j)) & 1) { da += le[j]; db += le[7 + j]; }
  }
  EA[(size_t)b * 128 + t] = expf(da - shi);
  EB[(size_t)b * 128 + t] = expf(db - slo);
  if (t == 0) atomicAdd(out, -(shi + slo));     // -EIV
}

// ------------- stage 3: pi[c,k], log_terms, piLT = pi*log_terms ------------
__global__ void __launch_bounds__(128)
k_pi(const float* __restrict__ beta_0, const float* __restrict__ beta,
     float* __restrict__ pi, float* __restrict__ piLT) {
  __shared__ float bs[CC * MM];
  __shared__ float b0s[CC];
  const int t = threadIdx.x;
  for (int i = t; i < CC * MM; i += 128) bs[i] = beta[i];
  if (t < CC) b0s[t] = beta_0[t];
  __syncthreads();

  const int k = blockIdx.x * 128 + t;
  float sg[MM];
#pragma unroll
  for (int j = 0; j < MM; ++j) sg[j] = ((k >> (MM - 1 - j)) & 1) ? 1.f : -1.f;

  float mx = -3.4e38f;
  for (int c = 0; c < CC; ++c) {
    float lp = b0s[c];
#pragma unroll
    for (int j = 0; j < MM; ++j) lp += bs[c * MM + j] * sg[j];
    mx = fmaxf(mx, lp);
  }
  float sum = 0.f;
  for (int c = 0; c < CC; ++c) {
    float lp = b0s[c];
#pragma unroll
    for (int j = 0; j < MM; ++j) lp += bs[c * MM + j] * sg[j];
    sum += expf(lp - mx);
  }
  const float lt = mx + logf(sum);
  const float inv = 1.f / sum;
  for (int c = 0; c < CC; ++c) {
    float lp = b0s[c];
#pragma unroll
    for (int j = 0; j < MM; ++j) lp += bs[c * MM + j] * sg[j];
    float p = expf(lp - mx) * inv;
    pi[(size_t)c * KK + k] = p;          // coalesced over k
    piLT[(size_t)c * KK + k] = p * lt;
  }
}

// --------------------- stage 4: group samples by class ---------------------
__global__ void k_hist(const int* __restrict__ y, int* __restrict__ counts) {
  int b = blockIdx.x * blockDim.x + threadIdx.x;
  if (b < BB) atomicAdd(&counts[y[b]], 1);
}
__global__ void k_scan(const int* __restrict__ counts, int* __restrict__ start,
                       int* __restrict__ cursor) {
  if (threadIdx.x == 0 && blockIdx.x == 0) {
    int run = 0;
    for (int c = 0; c < CC; ++c) { start[c] = run; cursor[c] = run; run += counts[c]; }
  }
}
__global__ void k_scatter(const int* __restrict__ y, int* __restrict__ cursor,
                          int* __restrict__ order) {
  int b = blockIdx.x * blockDim.x + threadIdx.x;
  if (b < BB) {
    int pos = atomicAdd(&cursor[y[b]], 1);
    order[pos] = b;
  }
}

// ----------- stage 5: per-class WMMA bilinear forms + finalize -------------
// One block per class. LDS: V_c (128x132), V_c*LT (128x132), sample tiles and
// u/uL/v outputs (stride 132). ~180 KB -> fits the 320 KB CDNA5 WGP LDS.
__global__ void __launch_bounds__(256)
k_classes(const float* __restrict__ EA, const float* __restrict__ EB,
          const float* __restrict__ pi, const float* __restrict__ piLT,
          const float* __restrict__ le_pad, const int* __restrict__ order,
          const int* __restrict__ start, const int* __restrict__ counts,
          const float* __restrict__ beta_0, const float* __restrict__ beta,
          float* __restrict__ out) {
  extern __shared__ __align__(16) char smem[];
  float* Vs   = (float*)smem;                    // [128][132]
  float* VLTs = Vs + 128 * 132;                  // [128][132]
  float* sEB  = VLTs + 128 * 132;                // [16][132]
  float* sEA  = sEB + 16 * 132;                  // [16][132]
  float* u_s  = sEA + 16 * 132;                  // [16][132]
  float* uL_s = u_s + 16 * 132;                  // [16][132]
  float* v_s  = uL_s + 16 * 132;                 // [16][132]
  float* le_s = v_s + 16 * 132;                  // [16][16]
  int*   bval = (int*)(le_s + 16 * 16);          // [16] validity

  const int c = blockIdx.x;
  const int n_c = counts[c];
  if (n_c == 0) return;
  const int st = start[c];
  const int t = threadIdx.x;
  const int wave = t >> 5;        // 8 waves
  const int lane = t & 31;

  // stage V_c and V_c*LT once per class, float4 (L2 resident, coalesced)
  {
    const float4* pv = (const float4*)(pi + (size_t)c * KK);
    const float4* lv = (const float4*)(piLT + (size_t)c * KK);
    for (int i = 0; i < 16; ++i) {
      int e = t + i * 256;                       // float4 index, 4096 total
      int hi = e >> 5, l4 = e & 31;
      *(float4*)(Vs + hi * 132 + l4 * 4)   = pv[e];
      *(float4*)(VLTs + hi * 132 + l4 * 4) = lv[e];
    }
  }
  __syncthreads();

  const int nch = (n_c + 15) >> 4;
  for (int ch = 0; ch < nch; ++ch) {
    // ---- load 16-sample tiles of EA / EB / le (float4) ----
    {
      int r = t >> 4, g = t & 15;
      int s = ch * 16 + r;
      int valid = (s < n_c);
      int b = valid ? order[st + s] : 0;
      if (g == 0) bval[r] = valid;
      const float4 z = make_float4(0.f, 0.f, 0.f, 0.f);
#pragma unroll
      for (int i = 0; i < 2; ++i) {
        int c4 = g + i * 16;                     // 0..31
        float4 vb = valid ? ((const float4*)(EB + (size_t)b * 128))[c4] : z;
        float4 va = valid ? ((const float4*)(EA + (size_t)b * 128))[c4] : z;
        *(float4*)(sEB + r * 132 + c4 * 4) = vb;
        *(float4*)(sEA + r * 132 + c4 * 4) = va;
      }
      le_s[r * 16 + g] = valid ? le_pad[(size_t)b * LEP + g] : 0.f;
    }
    __syncthreads();

    // ---- three 16x128 (K=128) fp32 WMMA GEMMs; wave w owns columns w*16.. ----
    {
      const int mrow = lane & 15;
      const int off0 = (lane < 16) ? 0 : 2;
      const int ncol = wave * 16 + (lane & 15);
      v8f au = {0,0,0,0,0,0,0,0}, al = {0,0,0,0,0,0,0,0}, av = {0,0,0,0,0,0,0,0};
#pragma unroll 4
      for (int kk = 0; kk < 128; kk += 4) {
        v2f a, b;
        // u / uL: A = EB tile, B[k][n] = V[n][k]  (n = hi, k = lo)
        a.x = sEB[mrow * 132 + kk + off0];
        a.y = sEB[mrow * 132 + kk + off0 + 1];
        b.x = Vs[ncol * 132 + kk + off0];
        b.y = Vs[ncol * 132 + kk + off0 + 1];
        au = wmma_f32_16x16x4(a, b, au);
        b.x = VLTs[ncol * 132 + kk + off0];
        b.y = VLTs[ncol * 132 + kk + off0 + 1];
        al = wmma_f32_16x16x4(a, b, al);
        // v: A = EA tile, B[k][n] = V[k][n]  (k = hi, n = lo)
        a.x = sEA[mrow * 132 + kk + off0];
        a.y = sEA[mrow * 132 + kk + off0 + 1];
        b.x = Vs[(kk + off0) * 132 + ncol];
        b.y = Vs[(kk + off0 + 1) * 132 + ncol];
        av = wmma_f32_16x16x4(a, b, av);
      }
#pragma unroll
      for (int r = 0; r < 8; ++r) {
        int mr = (lane < 16) ? r : (r + 8);
        u_s[mr * 132 + ncol]  = au[r];
        uL_s[mr * 132 + ncol] = al[r];
        v_s[mr * 132 + ncol]  = av[r];
      }
    }
    __syncthreads();

    // ---- per-sample finalization (16 lanes active; trivial work) ----
    if (t < 16 && bval[t] && (ch * 16 + t) < n_c) {
      const int s = t;
      float denom = 0.f, T = 0.f;
      float N[MM];
#pragma unroll
      for (int j = 0; j < MM; ++j) N[j] = 0.f;
      for (int h = 0; h < 128; ++h) {
        float ea = sEA[s * 132 + h];
        float eu = ea * u_s[s * 132 + h];
        denom += eu;
        T += ea * uL_s[s * 132 + h];
#pragma unroll
        for (int j = 0; j < 7; ++j)
          if ((h >> (6 - j)) & 1) N[j] += eu;
      }
      for (int l = 0; l < 128; ++l) {
        float ev = sEB[s * 132 + l] * v_s[s * 132 + l];
#pragma unroll
        for (int j = 7; j < MM; ++j)
          if ((l >> (MM - 1 - j)) & 1) N[j] += ev;
      }
      const float inv = 1.f / denom;
      float EI = beta_0[c];
      float EIII = 0.f;
#pragma unroll
      for (int j = 0; j < MM; ++j) {
        float E = N[j] * inv;
        EI += (2.f * E - 1.f) * beta[c * MM + j];
        EIII += E * le_s[s * 16 + j];
      }
      const float EII = T * inv;
      atomicAdd(out, EI - EII + EIII);
    }
    __syncthreads();
  }
}

// ------------------------------- launcher ----------------------------------
extern "C" void kernel_launch(void* const* d_in, const int* in_sizes, int n_in,
                              void* d_out, int out_size, void* d_ws, size_t ws_size,
                              hipStream_t stream) {
  const float* x_in    = (const float*)d_in[0];
  const int*   y_in    = (const int*)d_in[1];
  const float* alpha_0 = (const float*)d_in[2];
  const float* alpha   = (const float*)d_in[3];
  const float* beta_0  = (const float*)d_in[4];
  const float* beta    = (const float*)d_in[5];
  float* out = (float*)d_out;

  char* ws = (char*)d_ws;
  float* le_pad = (float*)(ws + 0);                    //  256 KB
  float* EA     = (float*)(ws + 262144);               //    2 MB
  float* EB     = (float*)(ws + 2359296);              //    2 MB
  float* pi     = (float*)(ws + 4456448);              //  6.4 MB
  float* piLT   = (float*)(ws + 11010048);             //  6.4 MB
  int*   counts = (int*)  (ws + 17563648);
  int*   cstart = (int*)  (ws + 17564160);
  int*   cursor = (int*)  (ws + 17564672);
  int*   order  = (int*)  (ws + 17565184);             //   16 KB

  k_init<<<(BB * LEP + 255) / 256, 256, 0, stream>>>(le_pad, alpha_0, counts, out);
  k_le_gemm<<<dim3(BB / 16, KSPLIT), 32, 0, stream>>>(x_in, alpha, le_pad);
  k_eab<<<BB, 128, 0, stream>>>(le_pad, EA, EB, out);
  k_pi<<<KK / 128, 128, 0, stream>>>(beta_0, beta, pi, piLT);
  k_hist<<<BB / 256, 256, 0, stream>>>(y_in, counts);
  k_scan<<<1, 32, 0, stream>>>(counts, cstart, cursor);
  k_scatter<<<BB / 256, 256, 0, stream>>>(y_in, cursor, order);

  const size_t shmem = (2 * 128 * 132 + 5 * 16 * 132 + 16 * 16) * sizeof(float)
                       + 16 * sizeof(int);
  k_classes<<<CC, 256, shmem, stream>>>(EA, EB, pi, piLT, le_pad, order,
                                        cstart, counts, beta_0, beta, out);
}